// _LoRA_qkv_timm_eval_73134703116333
// MI455X (gfx1250) — compile-verified
//
#include <hip/hip_runtime.h>
#include <math.h>

// ---------------------------------------------------------------------------
// LoRA-QKV (eval) for MI455X / gfx1250, wave32, WMMA f32 16x16x4.
//
//   out[m, d] = sum_c x[m,c] * W_qkv[d,c] + b[d]
//   out[m, 0:768]      += sum_{t,r} (x @ A_q[t]^T)[m,r] * scale_q[t] * B_q[t,d,r]
//   out[m, 1536:2304]  += same with (A_v, B_v, scale_v)
//
// scale_t = softmax(logits/0.5)_t / (||A_t||_F * ||B_t||_F)
//
// Strategy: fold scale into W2[d, t*16+r] = scale_t * B[t,d,r]  (prep kernel),
// then everything is a row-major NT GEMM  Out = X[M,K] @ W[N,K]^T (+bias).
//
// GEMM: 256 threads = 8 waves (4 in M x 2 in N). Wave tile 32 x (16*NSUB):
//   NSUB=4 -> block 128x128 (main + lora-up GEMMs), 8 WMMA per 6 ds_load_b64
//   NSUB=1 -> block 128x32  (N=160 U GEMMs)
// ---------------------------------------------------------------------------

typedef __attribute__((ext_vector_type(2))) float v2f;
typedef __attribute__((ext_vector_type(8))) float v8f;

#define TAU_INV 2.0f          // 1 / 0.5
#define T_TASKS 10
#define C_DIM   768
#define R_DIM   16
#define TR_DIM  (T_TASKS * R_DIM)      // 160
#define M_ROWS  6272                   // 32 * 196
#define D_OUT   2304

// ---------------------------------------------------------------------------
// prep: per branch (blockIdx.x = 0 -> q, 1 -> v) compute gate softmax,
// Frobenius norms, and emit W2[768][160] = scale_t * B[t, d, r].
// ---------------------------------------------------------------------------
__global__ __launch_bounds__(256) void prep_kernel(
    const float* __restrict__ A_q, const float* __restrict__ B_q,
    const float* __restrict__ g_q,
    const float* __restrict__ A_v, const float* __restrict__ B_v,
    const float* __restrict__ g_v,
    float* __restrict__ W2_base)
{
    const float* A  = blockIdx.x ? A_v : A_q;
    const float* Bm = blockIdx.x ? B_v : B_q;
    const float* g  = blockIdx.x ? g_v : g_q;
    float* Wd = W2_base + (size_t)blockIdx.x * (C_DIM * TR_DIM);

    __shared__ float red[256];
    __shared__ float nA[T_TASKS], nB[T_TASKS], scale[T_TASKS];
    const int tid = threadIdx.x;

    for (int t = 0; t < T_TASKS; ++t) {
        const float* At = A  + (size_t)t * R_DIM * C_DIM;
        const float* Bt = Bm + (size_t)t * C_DIM * R_DIM;
        float sa = 0.f, sb = 0.f;
        for (int i = tid; i < R_DIM * C_DIM; i += 256) {
            float xa = At[i]; sa += xa * xa;
            float xb = Bt[i]; sb += xb * xb;
        }
        red[tid] = sa; __syncthreads();
        for (int s = 128; s > 0; s >>= 1) {
            if (tid < s) red[tid] += red[tid + s];
            __syncthreads();
        }
        if (tid == 0) nA[t] = sqrtf(red[0]);
        __syncthreads();
        red[tid] = sb; __syncthreads();
        for (int s = 128; s > 0; s >>= 1) {
            if (tid < s) red[tid] += red[tid + s];
            __syncthreads();
        }
        if (tid == 0) nB[t] = sqrtf(red[0]);
        __syncthreads();
    }

    if (tid == 0) {
        float m = -3.0e38f;
        for (int t = 0; t < T_TASKS; ++t) m = fmaxf(m, g[t] * TAU_INV);
        float e[T_TASKS]; float den = 0.f;
        for (int t = 0; t < T_TASKS; ++t) { e[t] = __expf(g[t] * TAU_INV - m); den += e[t]; }
        for (int t = 0; t < T_TASKS; ++t) scale[t] = e[t] / (den * nA[t] * nB[t]);
    }
    __syncthreads();

    // W2[d][t*16 + r] = scale[t] * B[t][d][r]
    for (int i = tid; i < C_DIM * TR_DIM; i += 256) {
        int d  = i / TR_DIM;
        int tr = i - d * TR_DIM;
        int t  = tr >> 4;
        int r  = tr & 15;
        Wd[i] = scale[t] * Bm[(size_t)t * C_DIM * R_DIM + (size_t)d * R_DIM + r];
    }
}

// ---------------------------------------------------------------------------
// gemm_nt_wmma<NSUB>: Out[rowBase..+127, colOff+colBase..+32*NSUB-1] (=|+=)
//   = X[128,K] @ W[32*NSUB,K]^T (+ bias)
// Waves: wave = tid>>5; waveM = wave>>1 (0..3, 32 rows each), waveN = wave&1.
// Wave tile: 32 rows x 16*NSUB cols -> acc[2][NSUB] of v8f.
// Requires: M % 128 == 0, N % (32*NSUB) == 0, K % 32 == 0.
// ---------------------------------------------------------------------------
template <int NSUB>
__global__ __launch_bounds__(256) void gemm_nt_wmma(
    const float* __restrict__ X, const float* __restrict__ W,
    const float* __restrict__ bias, float* __restrict__ Out,
    int K, int lda, int ldb, int ldc, int colOff, int accum)
{
    constexpr int BN = 32 * NSUB;       // block column tile
    __shared__ float As[128][36];       // +4 pad: 16B-aligned, bank-spread
    __shared__ float Bs[BN][36];

    const int tid   = threadIdx.x;
    const int lane  = tid & 31;
    const int wave  = tid >> 5;
    const int waveM = wave >> 1;        // 0..3
    const int waveN = wave & 1;         // 0..1
    const int half  = lane >> 4;        // 0/1 (K-pair / row-half select)
    const int l16   = lane & 15;
    const int rowBase = blockIdx.y * 128;
    const int colBase = blockIdx.x * BN;

    v8f acc[2][NSUB];
#pragma unroll
    for (int i = 0; i < 2; ++i)
#pragma unroll
        for (int j = 0; j < NSUB; ++j) acc[i][j] = (v8f){};

    for (int kb = 0; kb < K; kb += 32) {
        // ---- stage A tile: 128 rows x 32 k-floats (4 float4 per thread) ----
#pragma unroll
        for (int i = 0; i < 4; ++i) {
            int idx = tid + i * 256;
            int r   = idx >> 3;
            int kv  = (idx & 7) * 4;
            *(float4*)&As[r][kv] =
                *(const float4*)(X + (size_t)(rowBase + r) * lda + kb + kv);
        }
        // ---- stage B tile: BN rows x 32 k-floats (BN/32 float4 per thread) --
#pragma unroll
        for (int i = 0; i < BN / 32; ++i) {
            int idx = tid + i * 256;
            int r   = idx >> 3;
            int kv  = (idx & 7) * 4;
            *(float4*)&Bs[r][kv] =
                *(const float4*)(W + (size_t)(colBase + r) * ldb + kb + kv);
        }
        // prefetch next A slab while we compute on this one
        if (kb + 32 < K)
            __builtin_prefetch(X + (size_t)(rowBase + (tid >> 1)) * lda + kb + 32, 0, 1);
        __syncthreads();

        // ---- 8 K-steps; per step: 2 A frags + NSUB B frags -> 2*NSUB WMMA --
#pragma unroll
        for (int ks = 0; ks < 32; ks += 4) {
            v2f a[2];
#pragma unroll
            for (int i = 0; i < 2; ++i)
                a[i] = *(const v2f*)&As[waveM * 32 + i * 16 + l16][ks + half * 2];
            v2f b[NSUB];
#pragma unroll
            for (int j = 0; j < NSUB; ++j)
                b[j] = *(const v2f*)&Bs[waveN * (16 * NSUB) + j * 16 + l16][ks + half * 2];
#pragma unroll
            for (int i = 0; i < 2; ++i)
#pragma unroll
                for (int j = 0; j < NSUB; ++j)
                    acc[i][j] = __builtin_amdgcn_wmma_f32_16x16x4_f32(
                        false, a[i], false, b[j], (short)0, acc[i][j], false, false);
        }
        __syncthreads();
    }

    // ---- epilogue: D element g -> row = g + 8*half, col = l16 ----
#pragma unroll
    for (int i = 0; i < 2; ++i) {
        const int rowTop = rowBase + waveM * 32 + i * 16 + half * 8;
#pragma unroll
        for (int j = 0; j < NSUB; ++j) {
            const int col = colOff + colBase + waveN * (16 * NSUB) + j * 16 + l16;
            const float bj = bias ? bias[col] : 0.f;
#pragma unroll
            for (int g = 0; g < 8; ++g) {
                size_t o = (size_t)(rowTop + g) * ldc + col;
                if (accum) Out[o] += acc[i][j][g];
                else       Out[o]  = acc[i][j][g] + bj;
            }
        }
    }
}

// ---------------------------------------------------------------------------
extern "C" void kernel_launch(void* const* d_in, const int* in_sizes, int n_in,
                              void* d_out, int out_size, void* d_ws, size_t ws_size,
                              hipStream_t stream)
{
    const float* x    = (const float*)d_in[0];   // [6272, 768]
    const float* Wqkv = (const float*)d_in[1];   // [2304, 768]
    const float* bqkv = (const float*)d_in[2];   // [2304]
    const float* A_q  = (const float*)d_in[3];   // [10,16,768] -> flat [160,768]
    const float* B_q  = (const float*)d_in[4];   // [10,768,16]
    const float* A_v  = (const float*)d_in[5];
    const float* B_v  = (const float*)d_in[6];
    const float* g_q  = (const float*)d_in[7];   // [10]
    const float* g_v  = (const float*)d_in[8];

    float* out = (float*)d_out;                  // [6272, 2304]
    float* ws  = (float*)d_ws;

    float* W2 = ws;                                   // 2 * 768*160
    float* Uq = ws + 2 * (size_t)(C_DIM * TR_DIM);    // 6272*160
    float* Uv = Uq + (size_t)M_ROWS * TR_DIM;

    // 1) gates / norms / folded up-projection weights
    prep_kernel<<<dim3(2), 256, 0, stream>>>(A_q, B_q, g_q, A_v, B_v, g_v, W2);

    const dim3 blk(256);
    const int  gy = M_ROWS / 128;   // 49

    // 2) U_q = x @ A_qcat^T   [6272,160]   (narrow-N variant)
    gemm_nt_wmma<1><<<dim3(TR_DIM / 32, gy), blk, 0, stream>>>(
        x, A_q, nullptr, Uq, C_DIM, C_DIM, C_DIM, TR_DIM, 0, 0);
    // 3) U_v = x @ A_vcat^T
    gemm_nt_wmma<1><<<dim3(TR_DIM / 32, gy), blk, 0, stream>>>(
        x, A_v, nullptr, Uv, C_DIM, C_DIM, C_DIM, TR_DIM, 0, 0);
    // 4) out = x @ W_qkv^T + b   [6272,2304]  (wide 128x128 tiles)
    gemm_nt_wmma<4><<<dim3(D_OUT / 128, gy), blk, 0, stream>>>(
        x, Wqkv, bqkv, out, C_DIM, C_DIM, C_DIM, D_OUT, 0, 0);
    // 5) out[:, 0:768]     += U_q @ W2_q^T
    gemm_nt_wmma<4><<<dim3(C_DIM / 128, gy), blk, 0, stream>>>(
        Uq, W2, nullptr, out, TR_DIM, TR_DIM, TR_DIM, D_OUT, 0, 1);
    // 6) out[:, 1536:2304] += U_v @ W2_v^T
    gemm_nt_wmma<4><<<dim3(C_DIM / 128, gy), blk, 0, stream>>>(
        Uv, W2 + (size_t)(C_DIM * TR_DIM), nullptr, out,
        TR_DIM, TR_DIM, TR_DIM, D_OUT, 2 * C_DIM, 1);
}